// HierarchyGAT_85237920957041
// MI455X (gfx1250) — compile-verified
//
#include <hip/hip_runtime.h>
#include <hip/hip_bf16.h>

// ---------------------------------------------------------------------------
// HierarchyGAT pipeline for MI455X (gfx1250, wave32)
//   L1: H = X@W1 ; GAT edge softmax ; OUT = segsum(alpha*H[src]) ; relu(+b1)
//   L2: same with W2
//   MLP: gather docs -> relu(D@Wm1+bm1) -> softmax(Dh@Wm2+bm2)
// Dense GEMMs use V_WMMA_F32_16X16X4_F32 (full fp32 precision); each wave
// drives two independent 16x16 accumulator chains for WMMA pipelining.
// ---------------------------------------------------------------------------

#define N_NODES   50000
#define N_EDGES   800000
#define D_HID     256
#define N_DOCS    20000
#define MLP_HID   64
#define N_CLASSES 20

typedef __attribute__((ext_vector_type(2))) float v2f;
typedef __attribute__((ext_vector_type(8))) float v8f;

// ---------------------------------------------------------------------------
// fp32 WMMA GEMM: C[M,N] = A[M,K] @ B[K,N]  (+bias, +relu optional)
// block = 64 threads (2 waves). Block tile: 16 (M) x 64 (N).
// Each wave owns a 16x32 strip = two independent 16x16 WMMA accumulators,
// so each A fragment is reused twice and the two chains pipeline in the XDL.
// K staged through LDS in chunks of 32; next chunk prefetched into L2.
// Requires: M%16==0, N%64==0, K%32==0. (50000/20000 x 256/64 all satisfy.)
// ---------------------------------------------------------------------------
__global__ __launch_bounds__(64) void gemm_wmma_f32(
    const float* __restrict__ A, const float* __restrict__ B,
    float* __restrict__ C, int M, int N, int K,
    const float* __restrict__ bias, int doRelu)
{
    __shared__ float As[16][32];  // M x Kc
    __shared__ float Bs[32][64];  // Kc x Nt

    const int tid  = threadIdx.x;
    const int wave = tid >> 5;          // 0..1 -> which 32-col half of N tile
    const int lane = tid & 31;
    const int m0 = blockIdx.x * 16;
    const int n0 = blockIdx.y * 64;

    v8f acc0 = {};
    v8f acc1 = {};

    const int mrow  = lane & 15;
    const int khalf = (lane >> 4) << 1;           // 0 or 2 (A/B VGPR K split)
    const int ncol0 = (lane & 15) + (wave << 5);  // col within 64-wide N tile
    const int ncol1 = ncol0 + 16;

    // Per-thread staging addresses
    const int ar = tid >> 2;            // A row 0..15
    const int ac = (tid & 3) * 8;       // A col 0,8,16,24
    const int br = tid >> 1;            // B row 0..31
    const int bc = (tid & 1) * 32;      // B col 0 or 32

    for (int k0 = 0; k0 < K; k0 += 32) {
        // Stage A chunk 16x32 (512 floats, 8 per thread, float4 coalesced)
        {
            const float* s = A + (size_t)(m0 + ar) * K + k0 + ac;
            *(float4*)&As[ar][ac]     = *(const float4*)(s);
            *(float4*)&As[ar][ac + 4] = *(const float4*)(s + 4);
        }
        // Stage B chunk 32x64 (2048 floats, 32 per thread)
        {
            const float* s = B + (size_t)(k0 + br) * N + n0 + bc;
            #pragma unroll
            for (int j = 0; j < 8; ++j)
                *(float4*)&Bs[br][bc + 4 * j] = *(const float4*)(s + 4 * j);
        }
        // Prefetch next K chunk into L2 while this chunk computes
        if (k0 + 32 < K) {
            __builtin_prefetch(A + (size_t)(m0 + ar) * K + k0 + 32 + ac, 0, 3);
            __builtin_prefetch(B + (size_t)(k0 + 32 + br) * N + n0 + bc, 0, 3);
        }
        __syncthreads();

        #pragma unroll
        for (int kk = 0; kk < 32; kk += 4) {
            v2f a, b0, b1;
            // A 16x4 layout: lanes 0-15 hold K=0,1 ; lanes 16-31 hold K=2,3
            a.x = As[mrow][kk + khalf];
            a.y = As[mrow][kk + khalf + 1];
            // B 4x16 layout (rows striped across lanes, symmetric K split)
            b0.x = Bs[kk + khalf][ncol0];
            b0.y = Bs[kk + khalf + 1][ncol0];
            b1.x = Bs[kk + khalf][ncol1];
            b1.y = Bs[kk + khalf + 1][ncol1];
            acc0 = __builtin_amdgcn_wmma_f32_16x16x4_f32(
                false, a, false, b0, (short)0, acc0, false, false);
            acc1 = __builtin_amdgcn_wmma_f32_16x16x4_f32(
                false, a, false, b1, (short)0, acc1, false, false);
        }
        __syncthreads();
    }

    // C/D 16x16 layout: VGPR r -> row r (lanes 0-15) / row 8+r (lanes 16-31)
    const int colg0 = n0 + ncol0;
    const int colg1 = n0 + ncol1;
    const float bv0 = bias ? bias[colg0] : 0.0f;
    const float bv1 = bias ? bias[colg1] : 0.0f;
    #pragma unroll
    for (int r = 0; r < 8; ++r) {
        const int rowg = m0 + r + ((lane >> 4) << 3);
        float v0 = acc0[r] + bv0;
        float v1 = acc1[r] + bv1;
        if (doRelu) { v0 = fmaxf(v0, 0.0f); v1 = fmaxf(v1, 0.0f); }
        C[(size_t)rowg * N + colg0] = v0;
        C[(size_t)rowg * N + colg1] = v1;
    }
}

// ---------------------------------------------------------------------------
// Per-node attention dots: es[n] = H[n,:] . a_s ; ed[n] = H[n,:] . a_d
// One wave32 per node, shuffle reduction.
// ---------------------------------------------------------------------------
__global__ __launch_bounds__(256) void node_dots(
    const float* __restrict__ H, const float* __restrict__ as_,
    const float* __restrict__ ad_, float* __restrict__ es,
    float* __restrict__ ed, int n)
{
    const int node = (int)((blockIdx.x * blockDim.x + threadIdx.x) >> 5);
    const int lane = threadIdx.x & 31;
    if (node >= n) return;
    const float* row = H + (size_t)node * D_HID;
    float s = 0.0f, d = 0.0f;
    #pragma unroll
    for (int j = 0; j < D_HID / 32; ++j) {
        const int i = lane + j * 32;
        const float h = row[i];
        s += h * as_[i];
        d += h * ad_[i];
    }
    #pragma unroll
    for (int off = 16; off; off >>= 1) {
        s += __shfl_down(s, off, 32);
        d += __shfl_down(d, off, 32);
    }
    if (lane == 0) { es[node] = s; ed[node] = d; }
}

__global__ void fill_f32(float* __restrict__ p, float v, size_t n)
{
    const size_t i = (size_t)blockIdx.x * blockDim.x + threadIdx.x;
    if (i < n) p[i] = v;
}

// Monotonic float atomic max via signed-max / unsigned-min pair.
__device__ inline void atomicMaxF(float* addr, float val)
{
    if (val >= 0.0f) atomicMax((int*)addr, __float_as_int(val));
    else             atomicMin((unsigned int*)addr, __float_as_uint(val));
}

// Pass 1: e = leaky_relu(es[src]+ed[dst]); segment max into mx[dst]
__global__ __launch_bounds__(256) void edge_score(
    const int* __restrict__ src, const int* __restrict__ dst,
    const float* __restrict__ es, const float* __restrict__ ed,
    float* __restrict__ e, float* __restrict__ mx, int nE)
{
    const int i = blockIdx.x * blockDim.x + threadIdx.x;
    if (i >= nE) return;
    float v = es[src[i]] + ed[dst[i]];
    v = (v > 0.0f) ? v : 0.2f * v;
    e[i] = v;
    atomicMaxF(&mx[dst[i]], v);
}

// Pass 2: e <- exp(e - mx[dst]); den[dst] += e
__global__ __launch_bounds__(256) void edge_exp(
    const int* __restrict__ dst, const float* __restrict__ mx,
    float* __restrict__ e, float* __restrict__ den, int nE)
{
    const int i = blockIdx.x * blockDim.x + threadIdx.x;
    if (i >= nE) return;
    const int d = dst[i];
    const float ex = __expf(e[i] - mx[d]);
    e[i] = ex;
    atomicAdd(&den[d], ex);
}

// Pass 3: OUT[dst,:] += (e/(den[dst]+eps)) * H[src,:]   (one wave per edge)
__global__ __launch_bounds__(256) void edge_aggregate(
    const int* __restrict__ src, const int* __restrict__ dst,
    const float* __restrict__ e, const float* __restrict__ den,
    const float* __restrict__ H, float* __restrict__ OUT, int nE)
{
    const int edge = (int)((blockIdx.x * blockDim.x + threadIdx.x) >> 5);
    const int lane = threadIdx.x & 31;
    if (edge >= nE) return;
    const int s = src[edge], d = dst[edge];
    const float alpha = e[edge] / (den[d] + 1e-16f);
    const float* hrow = H + (size_t)s * D_HID;
    float* orow = OUT + (size_t)d * D_HID;
    #pragma unroll
    for (int j = 0; j < D_HID / 32; ++j) {
        const int c = lane + j * 32;  // coalesced across the wave
        atomicAdd(&orow[c], alpha * hrow[c]);
    }
}

// X <- relu(X + b[col]) elementwise over [n, 256]
__global__ void bias_relu256(float* __restrict__ X, const float* __restrict__ b,
                             size_t n)
{
    const size_t i = (size_t)blockIdx.x * blockDim.x + threadIdx.x;
    if (i >= n) return;
    X[i] = fmaxf(X[i] + b[i & (D_HID - 1)], 0.0f);
}

__global__ void gather_rows256(const float* __restrict__ X,
                               const int* __restrict__ idx,
                               float* __restrict__ Y, size_t n)
{
    const size_t i = (size_t)blockIdx.x * blockDim.x + threadIdx.x;
    if (i >= n) return;
    const size_t r = i >> 8, c = i & (D_HID - 1);
    Y[i] = X[(size_t)idx[r] * D_HID + c];
}

// Final: logits = DH@Wm2 + bm2 ; softmax. One wave per doc (20 classes in lanes).
__global__ __launch_bounds__(256) void mlp2_softmax(
    const float* __restrict__ DH, const float* __restrict__ Wm2,
    const float* __restrict__ bm2, float* __restrict__ out, int nDocs)
{
    const int doc  = (int)((blockIdx.x * blockDim.x + threadIdx.x) >> 5);
    const int lane = threadIdx.x & 31;
    if (doc >= nDocs) return;
    const float* row = DH + (size_t)doc * MLP_HID;
    float logit = -3.0e38f;
    if (lane < N_CLASSES) {
        float acc = bm2[lane];
        #pragma unroll
        for (int k = 0; k < MLP_HID; ++k)
            acc += row[k] * Wm2[k * N_CLASSES + lane];
        logit = acc;
    }
    float mx = logit;
    #pragma unroll
    for (int off = 16; off; off >>= 1) mx = fmaxf(mx, __shfl_xor(mx, off, 32));
    const float ex = (lane < N_CLASSES) ? __expf(logit - mx) : 0.0f;
    float sum = ex;
    #pragma unroll
    for (int off = 16; off; off >>= 1) sum += __shfl_xor(sum, off, 32);
    if (lane < N_CLASSES) out[(size_t)doc * N_CLASSES + lane] = ex / sum;
}

// ---------------------------------------------------------------------------
extern "C" void kernel_launch(void* const* d_in, const int* in_sizes, int n_in,
                              void* d_out, int out_size, void* d_ws, size_t ws_size,
                              hipStream_t stream)
{
    (void)in_sizes; (void)n_in; (void)out_size; (void)ws_size;

    const float* x    = (const float*)d_in[0];
    const int*   ei   = (const int*)  d_in[1];
    const int*   doci = (const int*)  d_in[2];
    const float* W1   = (const float*)d_in[3];
    const float* a1s  = (const float*)d_in[4];
    const float* a1d  = (const float*)d_in[5];
    const float* b1   = (const float*)d_in[6];
    const float* W2   = (const float*)d_in[7];
    const float* a2s  = (const float*)d_in[8];
    const float* a2d  = (const float*)d_in[9];
    const float* b2   = (const float*)d_in[10];
    const float* Wm1  = (const float*)d_in[11];
    const float* bm1  = (const float*)d_in[12];
    const float* Wm2  = (const float*)d_in[13];
    const float* bm2  = (const float*)d_in[14];

    const int* src = ei;            // edge_index[0]
    const int* dst = ei + N_EDGES;  // edge_index[1]

    // Workspace carve-up (floats); all regions fully rewritten every call.
    float* H    = (float*)d_ws;                      // [N_NODES, 256]
    float* OUT  = H    + (size_t)N_NODES * D_HID;    // [N_NODES, 256]
    float* ES   = OUT  + (size_t)N_NODES * D_HID;    // [N_NODES]
    float* ED   = ES   + N_NODES;                    // [N_NODES]
    float* MX   = ED   + N_NODES;                    // [N_NODES]
    float* DEN  = MX   + N_NODES;                    // [N_NODES]
    float* EB   = DEN  + N_NODES;                    // [N_EDGES]
    float* DOCS = EB   + N_EDGES;                    // [N_DOCS, 256]
    float* DH   = DOCS + (size_t)N_DOCS * D_HID;     // [N_DOCS, 64]

    const dim3 gemmGridN(N_NODES / 16, D_HID / 64);      // 3125 x 4
    const dim3 gemmGridD(N_DOCS  / 16, MLP_HID / 64);    // 1250 x 1
    const size_t feat = (size_t)N_NODES * D_HID;         // 12.8M
    const int gFeat  = (int)((feat + 255) / 256);
    const int gNode  = (N_NODES + 255) / 256;
    const int gNodeW = (N_NODES + 7) / 8;                // wave-per-node
    const int gEdge  = (N_EDGES + 255) / 256;
    const int gEdgeW = (N_EDGES + 7) / 8;                // wave-per-edge
    const size_t docf = (size_t)N_DOCS * D_HID;
    const int gDocF  = (int)((docf + 255) / 256);
    const int gDocW  = (N_DOCS + 7) / 8;

    for (int layer = 0; layer < 2; ++layer) {
        const float* A   = (layer == 0) ? x   : OUT;  // layer input
        const float* W   = (layer == 0) ? W1  : W2;
        const float* as_ = (layer == 0) ? a1s : a2s;
        const float* ad_ = (layer == 0) ? a1d : a2d;
        const float* b   = (layer == 0) ? b1  : b2;

        // H = A @ W  (fp32 WMMA)
        gemm_wmma_f32<<<gemmGridN, 64, 0, stream>>>(A, W, H, N_NODES, D_HID,
                                                    D_HID, nullptr, 0);
        // attention scalars
        node_dots<<<gNodeW, 256, 0, stream>>>(H, as_, ad_, ES, ED, N_NODES);
        // reset segment buffers
        fill_f32<<<gNode, 256, 0, stream>>>(MX, -INFINITY, (size_t)N_NODES);
        fill_f32<<<gNode, 256, 0, stream>>>(DEN, 0.0f, (size_t)N_NODES);
        fill_f32<<<gFeat, 256, 0, stream>>>(OUT, 0.0f, feat);
        // segment softmax + weighted aggregation
        edge_score<<<gEdge, 256, 0, stream>>>(src, dst, ES, ED, EB, MX, N_EDGES);
        edge_exp<<<gEdge, 256, 0, stream>>>(dst, MX, EB, DEN, N_EDGES);
        edge_aggregate<<<gEdgeW, 256, 0, stream>>>(src, dst, EB, DEN, H, OUT,
                                                   N_EDGES);
        // OUT = relu(OUT + b)
        bias_relu256<<<gFeat, 256, 0, stream>>>(OUT, b, feat);
    }

    // MLP head
    gather_rows256<<<gDocF, 256, 0, stream>>>(OUT, doci, DOCS, docf);
    gemm_wmma_f32<<<gemmGridD, 64, 0, stream>>>(DOCS, Wm1, DH, N_DOCS, MLP_HID,
                                                D_HID, bm1, 1);
    mlp2_softmax<<<gDocW, 256, 0, stream>>>(DH, Wm2, bm2, (float*)d_out, N_DOCS);
}